// ConfusionAwareLoss_2989297238223
// MI455X (gfx1250) — compile-verified
//
#include <hip/hip_runtime.h>
#include <stdint.h>

#define NCLS            1000
#define ROW_BYTES       4000        // NCLS * 4
#define BUF_BYTES       4096        // padded row buffer in LDS
#define WAVES_PER_BLOCK 4
#define THREADS         (WAVES_PER_BLOCK * 32)
#define ROWS_PER_WAVE   8
#define TAIL_BASE       3584        // 7 * 512
#define TAIL_CLAMP      (ROW_BYTES - 16)   // 3984

typedef __attribute__((address_space(3))) unsigned char lds_uchar_t;

// ---- CDNA5 async global->LDS copy, GVS mode (saddr + vgpr offset + imm) ----
// Per CDNA5 ISA: the imm offset is added to both the LDS and the global address
// so one imm walks both addresses together. One instruction = 512 B per wave.
template <int IMM>
__device__ __forceinline__ void async_ld_b128(uint32_t lds_addr, uint32_t voff, uint64_t sbase) {
    asm volatile("global_load_async_to_lds_b128 %0, %1, %2 offset:%3"
                 :: "v"(lds_addr), "v"(voff), "s"(sbase), "i"(IMM)
                 : "memory");
}
__device__ __forceinline__ void wait_async_le8() {
    asm volatile("s_wait_asynccnt 8" ::: "memory");   // prev row done; next row may be in flight
}
__device__ __forceinline__ void wait_async_0() {
    asm volatile("s_wait_asynccnt 0" ::: "memory");
}

__global__ __launch_bounds__(THREADS)
void confusion_loss_rows_kernel(const float* __restrict__ outputs,
                                const int*   __restrict__ targets,
                                const float* __restrict__ penalty,
                                float*       __restrict__ partials) {
    __shared__ __align__(16) unsigned char smem[WAVES_PER_BLOCK * 2 * BUF_BYTES];
    __shared__ float wpart[WAVES_PER_BLOCK];

    const int lane  = threadIdx.x & 31;
    const int wave  = threadIdx.x >> 5;
    const long long gwave = (long long)blockIdx.x * WAVES_PER_BLOCK + wave;
    const long long row0  = gwave * ROWS_PER_WAVE;

    unsigned char* mybuf = smem + wave * 2 * BUF_BYTES;
    const uint32_t lds_base = (uint32_t)(size_t)(lds_uchar_t*)mybuf;

    // row-invariant pieces of the transfer addressing (computed once)
    const uint32_t lane16 = (uint32_t)(lane * 16);
    uint32_t gtail = TAIL_BASE + lane16;                 // t==7 global offset,
    if (gtail > TAIL_CLAMP) gtail = TAIL_CLAMP;          // clamped in-row (dups land in idx>=1000, masked)

    // ---- issue one row's 8 async b128 transfers into buffer b ----
    auto issue_row = [&](int r, int b) {
        const uint64_t sbase = (uint64_t)(const unsigned char*)(outputs + (size_t)(row0 + r) * NCLS);
        const uint32_t vlds  = lds_base + (uint32_t)b * BUF_BYTES + lane16;
        async_ld_b128<0 * 512>(vlds, lane16, sbase);
        async_ld_b128<1 * 512>(vlds, lane16, sbase);
        async_ld_b128<2 * 512>(vlds, lane16, sbase);
        async_ld_b128<3 * 512>(vlds, lane16, sbase);
        async_ld_b128<4 * 512>(vlds, lane16, sbase);
        async_ld_b128<5 * 512>(vlds, lane16, sbase);
        async_ld_b128<6 * 512>(vlds, lane16, sbase);
        async_ld_b128<0>(vlds + TAIL_BASE, gtail, sbase);   // tail: LDS unclamped, global clamped
    };

    // ---- consume one row from buffer b: returns base_loss * penalty ----
    auto compute_row = [&](int r, int b) -> float {
        const float* bf = (const float*)(mybuf + b * BUF_BYTES);

        // single LDS read of the row into registers (8 x float4 = 32 VGPRs)
        float4 rv[8];
#pragma unroll
        for (int t = 0; t < 8; ++t)
            rv[t] = *(const float4*)(bf + t * 128 + lane * 4);

        // pass 1 (regs): max + argmax, first-index tie-break like jnp.argmax
        float m  = -__builtin_inff();
        int   mi = NCLS;
#pragma unroll
        for (int t = 0; t < 8; ++t) {
            const int base = t * 128 + lane * 4;
            const float vals[4] = {rv[t].x, rv[t].y, rv[t].z, rv[t].w};
#pragma unroll
            for (int j = 0; j < 4; ++j) {
                const int idx = base + j;
                if (idx < NCLS) {
                    const float x = vals[j];
                    if (x > m || (x == m && idx < mi)) { m = x; mi = idx; }
                }
            }
        }
#pragma unroll
        for (int off = 16; off > 0; off >>= 1) {       // wave32 reduction
            const float om = __shfl_xor(m,  off, 32);
            const int   oi = __shfl_xor(mi, off, 32);
            if (om > m || (om == m && oi < mi)) { m = om; mi = oi; }
        }

        // pass 2 (regs): sum exp(x - m)
        float s = 0.0f;
#pragma unroll
        for (int t = 0; t < 8; ++t) {
            const int base = t * 128 + lane * 4;
            const float vals[4] = {rv[t].x, rv[t].y, rv[t].z, rv[t].w};
#pragma unroll
            for (int j = 0; j < 4; ++j) {
                const int idx = base + j;
                if (idx < NCLS) s += __expf(vals[j] - m);
            }
        }
#pragma unroll
        for (int off = 16; off > 0; off >>= 1) s += __shfl_xor(s, off, 32);

        const int   tgt = targets[row0 + r];
        const float xt  = bf[tgt];                                   // one LDS broadcast read
        const float base_loss = m + __logf(s) - xt;                  // logsumexp - x[target]
        const float pen = penalty[(size_t)tgt * NCLS + mi];          // 4 MB table, L2-resident
        return base_loss * pen;
    };

    // ---- double-buffered software pipeline over this wave's 8 rows ----
    float acc = 0.0f;
    issue_row(0, 0);
#pragma unroll
    for (int r = 0; r < ROWS_PER_WAVE; ++r) {
        if (r + 1 < ROWS_PER_WAVE) {
            issue_row(r + 1, (r + 1) & 1);
            wait_async_le8();
        } else {
            wait_async_0();
        }
        acc += compute_row(r, r & 1);
    }

    if (lane == 0) wpart[wave] = acc;
    __syncthreads();
    if (threadIdx.x == 0) {
        float s = 0.0f;
#pragma unroll
        for (int w = 0; w < WAVES_PER_BLOCK; ++w) s += wpart[w];
        partials[blockIdx.x] = s;
    }
}

// Deterministic final reduction: fixed single-block order, then mean.
__global__ __launch_bounds__(256)
void confusion_loss_reduce_kernel(const float* __restrict__ partials, int n,
                                  float* __restrict__ out, float inv_b) {
    __shared__ float sm[256];
    float s = 0.0f;
    for (int i = threadIdx.x; i < n; i += 256) s += partials[i];
    sm[threadIdx.x] = s;
    __syncthreads();
    for (int off = 128; off > 0; off >>= 1) {
        if (threadIdx.x < off) sm[threadIdx.x] += sm[threadIdx.x + off];
        __syncthreads();
    }
    if (threadIdx.x == 0) out[0] = sm[0] * inv_b;
}

extern "C" void kernel_launch(void* const* d_in, const int* in_sizes, int n_in,
                              void* d_out, int out_size, void* d_ws, size_t ws_size,
                              hipStream_t stream) {
    const float* outputs  = (const float*)d_in[0];   // [B, 1000] f32
    const int*   targets  = (const int*)  d_in[1];   // [B] i32
    const float* penalty  = (const float*)d_in[2];   // [1000, 1000] f32
    float*       out      = (float*)d_out;           // scalar f32
    float*       partials = (float*)d_ws;

    const int B       = in_sizes[1];                        // 131072
    const int nwaves  = B / ROWS_PER_WAVE;                  // 16384
    const int nblocks = nwaves / WAVES_PER_BLOCK;           // 4096

    confusion_loss_rows_kernel<<<nblocks, THREADS, 0, stream>>>(outputs, targets, penalty, partials);
    confusion_loss_reduce_kernel<<<1, 256, 0, stream>>>(partials, nblocks, out, 1.0f / (float)B);
}